// SAGDFNModel_41472204210852
// MI455X (gfx1250) — compile-verified
//
#include <hip/hip_runtime.h>
#include <hip/hip_bf16.h>

typedef __attribute__((ext_vector_type(2))) float v2f;
typedef __attribute__((ext_vector_type(4))) float v4f;
typedef __attribute__((ext_vector_type(8))) float v8f;

__device__ __forceinline__ v8f wmma_f32(v2f a, v2f b, v8f c) {
    // V_WMMA_F32_16X16X4_F32 : D = A(16x4) * B(4x16) + C(16x16), fp32
    return __builtin_amdgcn_wmma_f32_16x16x4_f32(false, a, false, b, (short)0, c, false, false);
}

// ---------------------------------------------------------------------------
// Node embedding MLP + q/k/node_hidden projections (tiny: ~30 MFLOP, VALU)
// e = relu(relu(feas@w3+b3)@w4+b4)@w5+b5 ; q=e@wq ; kk=e@wk ; nh=e@wnhp+bnhp
// ---------------------------------------------------------------------------
__global__ __launch_bounds__(256) void embed_kernel(
    const float* __restrict__ feas,
    const float* __restrict__ w3, const float* __restrict__ b3,
    const float* __restrict__ w4, const float* __restrict__ b4,
    const float* __restrict__ w5, const float* __restrict__ b5,
    const float* __restrict__ wq, const float* __restrict__ wk,
    const float* __restrict__ wnhp, const float* __restrict__ bnhp,
    float* __restrict__ q, float* __restrict__ kk, float* __restrict__ nh)
{
    __shared__ float s_w3[32 * 64], s_w4[64 * 64], s_w5[64 * 32];
    __shared__ float s_wq[32 * 64], s_wk[32 * 64], s_wn[32 * 64];
    __shared__ float s_b3[64], s_b4[64], s_b5[32], s_bn[64];
    int tid = threadIdx.x;
    for (int i = tid; i < 32 * 64; i += 256) {
        s_w3[i] = w3[i]; s_wq[i] = wq[i]; s_wk[i] = wk[i]; s_wn[i] = wnhp[i];
    }
    for (int i = tid; i < 64 * 64; i += 256) s_w4[i] = w4[i];
    for (int i = tid; i < 64 * 32; i += 256) s_w5[i] = w5[i];
    if (tid < 64) { s_b3[tid] = b3[tid]; s_b4[tid] = b4[tid]; s_bn[tid] = bnhp[tid]; }
    if (tid < 32) s_b5[tid] = b5[tid];
    __syncthreads();

    int n = blockIdx.x * 256 + tid;
    if (n >= 2000) return;

    float f[32], t1[64], t2[64], e3[32];
    for (int i = 0; i < 32; ++i) f[i] = feas[n * 32 + i];
    for (int j = 0; j < 64; ++j) {
        float s = s_b3[j];
        for (int i = 0; i < 32; ++i) s += f[i] * s_w3[i * 64 + j];
        t1[j] = fmaxf(s, 0.f);
    }
    for (int j = 0; j < 64; ++j) {
        float s = s_b4[j];
        for (int i = 0; i < 64; ++i) s += t1[i] * s_w4[i * 64 + j];
        t2[j] = fmaxf(s, 0.f);
    }
    for (int j = 0; j < 32; ++j) {
        float s = s_b5[j];
        for (int i = 0; i < 64; ++i) s += t2[i] * s_w5[i * 32 + j];
        e3[j] = s;
    }
    for (int j = 0; j < 64; ++j) {
        float sq = 0.f, sk = 0.f, sn = s_bn[j];
        for (int i = 0; i < 32; ++i) {
            float e = e3[i];
            sq += e * s_wq[i * 64 + j];
            sk += e * s_wk[i * 64 + j];
            sn += e * s_wn[i * 64 + j];
        }
        q[n * 64 + j]  = sq;
        kk[n * 64 + j] = sk;
        nh[n * 64 + j] = sn;
    }
}

// ---------------------------------------------------------------------------
// Attention scores + softmax -> w_adj (one wave = one node, 32 lanes = 32 nbrs)
// ---------------------------------------------------------------------------
__global__ __launch_bounds__(256) void attn_kernel(
    const float* __restrict__ q, const float* __restrict__ kk,
    const int* __restrict__ idx, float* __restrict__ w_adj)
{
    int lane = threadIdx.x & 31;
    int n = blockIdx.x * 8 + (threadIdx.x >> 5);   // 250 blocks * 8 = 2000 exact
    int j = idx[n * 32 + lane];
    const float* qr = q + (size_t)n * 64;
    const float* kr = kk + (size_t)j * 64;
    float s = 0.f;
    for (int c = 0; c < 64; ++c) s += qr[c] * kr[c];
    s *= 0.125f;                                    // 1/sqrt(64)
    float m = s;
    for (int o = 16; o >= 1; o >>= 1) m = fmaxf(m, __shfl_xor(m, o, 32));
    float ex = __expf(s - m);
    float sum = ex;
    for (int o = 16; o >= 1; o >>= 1) sum += __shfl_xor(sum, o, 32);
    w_adj[n * 32 + lane] = ex / sum;
}

// ---------------------------------------------------------------------------
// fused = x*w_in + b_in + node_hidden  -> h0[(b*12+t)*2000+n][c]
// ---------------------------------------------------------------------------
__global__ __launch_bounds__(256) void fuse_kernel(
    const float* __restrict__ inputs, const float* __restrict__ w_in,
    const float* __restrict__ b_in, const float* __restrict__ nh,
    float* __restrict__ h0)
{
    int i = blockIdx.x * 256 + threadIdx.x;        // 24000 blocks -> 6,144,000 exact
    int c  = i & 63;
    int rn = i >> 6;                               // bt*2000+n
    int n  = rn % 2000;
    int bt = rn / 2000;
    int b = bt / 12, t = bt % 12;
    float x = inputs[(t * 4 + b) * 2000 + n];
    h0[i] = x * w_in[c] + b_in[c] + nh[n * 64 + c];
}

// ---------------------------------------------------------------------------
// One SAGE layer, fully fused: gather+weighted agg (LDS tile) then
// h_out = relu(h @ W_self + b_self + agg @ W_neigh + b_neigh) via WMMA f32.
// Weights pre-packed in LDS as float4 {Wself[kb],Wself[kb+1],Wngh[kb],Wngh[kb+1]}
// per (k-pair, col) so one ds_load_b128 feeds both WMMA B operands directly.
// 4 waves/block, 1 wave = 16 rows (one bt, 16 consecutive n). 1500 blocks.
// ---------------------------------------------------------------------------
__global__ __launch_bounds__(128) void sage_kernel(
    const float* __restrict__ h_in, float* __restrict__ h_out,
    const float* __restrict__ w_self, const float* __restrict__ b_self,
    const float* __restrict__ w_neigh, const float* __restrict__ b_neigh,
    const float* __restrict__ w_adj, const int* __restrict__ idx)
{
    __shared__ v4f  s_wp[32 * 64];                 // 32 k-pairs x 64 cols, 32KB
    __shared__ float s_agg[4][16 * 64];            // per-wave agg tile, 16KB
    int tid = threadIdx.x;
    for (int i = tid; i < 2048; i += 128) {
        int p = i >> 6;                            // k-pair: rows {2p, 2p+1}
        int col = i & 63;
        v4f w;
        w.x = w_self [(2 * p)     * 64 + col];
        w.y = w_self [(2 * p + 1) * 64 + col];
        w.z = w_neigh[(2 * p)     * 64 + col];
        w.w = w_neigh[(2 * p + 1) * 64 + col];
        s_wp[i] = w;
    }
    __syncthreads();

    int widx = tid >> 5, lane = tid & 31;
    int wave = blockIdx.x * 4 + widx;              // 0..5999
    int bt = wave / 125;
    int n0 = (wave % 125) * 16;
    const float* hb = h_in + (size_t)bt * 2000 * 64;
    float* agg = s_agg[widx];

    // neighbor aggregation: lane owns channels {lane, lane+32}
    for (int i = 0; i < 16; ++i) {
        int n = n0 + i;
        float a0 = 0.f, a1 = 0.f;
        for (int m = 0; m < 32; ++m) {
            int j  = idx[n * 32 + m];              // broadcast load
            float w = w_adj[n * 32 + m];           // broadcast load
            const float* hr = hb + (size_t)j * 64;
            a0 += w * hr[lane];
            a1 += w * hr[lane + 32];
        }
        agg[i * 64 + lane]      = a0;
        agg[i * 64 + lane + 32] = a1;
    }

    int half = lane >> 4;                          // A/B K sub-select
    int r = lane & 15;                             // A row / B,C col
    const float* arow = hb + (size_t)(n0 + r) * 64;
    float* orow = h_out + ((size_t)bt * 2000 + n0) * 64;

    for (int ct = 0; ct < 4; ++ct) {
        int col = ct * 16 + r;
        v8f acc = {0.f, 0.f, 0.f, 0.f, 0.f, 0.f, 0.f, 0.f};
        for (int s = 0; s < 16; ++s) {
            int kb = 4 * s + 2 * half;
            float2 av = *reinterpret_cast<const float2*>(arow + kb);
            float2 gv = *reinterpret_cast<const float2*>(&agg[r * 64 + kb]);
            v4f bb = s_wp[(2 * s + half) * 64 + col];
            v2f a_self = {av.x, av.y};
            v2f a_agg  = {gv.x, gv.y};
            v2f b_self = {bb.x, bb.y};
            v2f b_ngh  = {bb.z, bb.w};
            acc = wmma_f32(a_self, b_self, acc);
            acc = wmma_f32(a_agg,  b_ngh,  acc);
        }
        float bias = b_self[col] + b_neigh[col];
        for (int j = 0; j < 8; ++j) {
            int row = j + 8 * half;                // C layout: vgpr j -> M=j / j+8
            float v = acc[j] + bias;
            orow[(size_t)row * 64 + col] = fmaxf(v, 0.f);
        }
    }
}

// ---------------------------------------------------------------------------
// (B,T,N,H) bt-major -> (T, B*N, H) t-major for the TCN
// ---------------------------------------------------------------------------
__global__ __launch_bounds__(256) void transpose_kernel(
    const float* __restrict__ mid, float* __restrict__ x0)
{
    int i = blockIdx.x * 256 + threadIdx.x;        // 6,144,000 exact
    int c  = i & 63;
    int rn = i >> 6;
    int n  = rn % 2000;
    int bt = rn / 2000;
    int b = bt / 12, t = bt % 12;
    x0[((size_t)(t * 8000 + b * 2000 + n)) * 64 + c] = mid[i];
}

// ---------------------------------------------------------------------------
// Dilated causal conv (K=2) as WMMA GEMM: y[t] = relu(b + x[t]@Wk1 + x[t-d]@Wk0)
// optional fused residual: y = relu(y + res). x layout (T=12, 8000, 64).
// Weights pre-packed in LDS as float4 {Wk1[kb],Wk1[kb+1],Wk0[kb],Wk0[kb+1]}.
// The 4 taps of one (col, k-pair) entry are contiguous in the (co,ci,k) global
// layout -> single b128 global load + swizzle per entry during staging.
// 1 wave = (t, 16 bn rows); 4 waves/block; 1500 blocks = 6000 wave-tiles.
// ---------------------------------------------------------------------------
__global__ __launch_bounds__(128) void tcn_conv_kernel(
    const float* __restrict__ x, float* __restrict__ y,
    const float* __restrict__ w, const float* __restrict__ bias,
    int dil, const float* __restrict__ res)
{
    __shared__ v4f s_wp[32 * 64];                  // 32 k-pairs x 64 cols, 32KB
    int tid = threadIdx.x;
    for (int i = tid; i < 2048; i += 128) {
        int p = i >> 6;                            // ci pair {2p, 2p+1}
        int col = i & 63;                          // co
        // w[(co*64+ci)*2 + k]: 4 contiguous floats = {ci=2p k0, k1, ci=2p+1 k0, k1}
        float4 wv = *reinterpret_cast<const float4*>(&w[(col * 64 + 2 * p) * 2]);
        v4f pk;
        pk.x = wv.y;                               // k=1 tap, ci=2p   (x[t])
        pk.y = wv.w;                               // k=1 tap, ci=2p+1
        pk.z = wv.x;                               // k=0 tap, ci=2p   (x[t-d])
        pk.w = wv.z;                               // k=0 tap, ci=2p+1
        s_wp[i] = pk;
    }
    __syncthreads();

    int widx = tid >> 5, lane = tid & 31;
    int wave = blockIdx.x * 4 + widx;              // 0..5999
    int t = wave % 12;
    int bn0 = (wave / 12) * 16;
    int half = lane >> 4;
    int r = lane & 15;
    const bool has_prev = (t >= dil);              // wave-uniform

    const float* a1row = x + ((size_t)t * 8000 + bn0 + r) * 64;
    const float* a0row = x + ((size_t)(has_prev ? (t - dil) : 0) * 8000 + bn0 + r) * 64;

    for (int ct = 0; ct < 4; ++ct) {
        int col = ct * 16 + r;
        v8f acc = {0.f, 0.f, 0.f, 0.f, 0.f, 0.f, 0.f, 0.f};
        if (has_prev) {
            for (int s = 0; s < 16; ++s) {
                int kb = 4 * s + 2 * half;
                float2 a1v = *reinterpret_cast<const float2*>(a1row + kb);
                float2 a0v = *reinterpret_cast<const float2*>(a0row + kb);
                v4f bb = s_wp[(2 * s + half) * 64 + col];
                v2f a1 = {a1v.x, a1v.y};
                v2f a0 = {a0v.x, a0v.y};
                v2f b1 = {bb.x, bb.y};
                v2f b0 = {bb.z, bb.w};
                acc = wmma_f32(a1, b1, acc);
                acc = wmma_f32(a0, b0, acc);
            }
        } else {
            for (int s = 0; s < 16; ++s) {
                int kb = 4 * s + 2 * half;
                float2 a1v = *reinterpret_cast<const float2*>(a1row + kb);
                v4f bb = s_wp[(2 * s + half) * 64 + col];
                v2f a1 = {a1v.x, a1v.y};
                v2f b1 = {bb.x, bb.y};
                acc = wmma_f32(a1, b1, acc);
            }
        }
        float bv = bias[col];
        for (int j = 0; j < 8; ++j) {
            int row = bn0 + j + 8 * half;
            float v = fmaxf(acc[j] + bv, 0.f);
            if (res) v = fmaxf(v + res[((size_t)t * 8000 + row) * 64 + col], 0.f);
            y[((size_t)t * 8000 + row) * 64 + col] = v;
        }
    }
}

// ---------------------------------------------------------------------------
// Gated head: gate=sigmoid(h_t@wg+bg); fused=g*h_t+(1-g)*h_s; out=fused@wo+bo
// outputs[hor, b, n]. One thread per (b,n) row.
// ---------------------------------------------------------------------------
__global__ __launch_bounds__(256) void head_kernel(
    const float* __restrict__ ht,                  // (12, 8000, 64), use t=11
    const float* __restrict__ mid,                 // (48, 2000, 64)
    const float* __restrict__ w_gate, const float* __restrict__ b_gate,
    const float* __restrict__ w_out, const float* __restrict__ b_out,
    float* __restrict__ out)                       // (12, 4, 2000)
{
    __shared__ float s_wo[64 * 12];
    __shared__ float s_wg[64];
    int tid = threadIdx.x;
    for (int i = tid; i < 768; i += 256) s_wo[i] = w_out[i];
    if (tid < 64) s_wg[tid] = w_gate[tid];
    __syncthreads();

    int i = blockIdx.x * 256 + tid;
    if (i >= 8000) return;
    int b = i / 2000, n = i % 2000;
    const float* htr = ht + ((size_t)11 * 8000 + i) * 64;
    const float* hsr = mid + ((size_t)(b * 12 + 11) * 2000 + n) * 64;

    float f[64];
    float g = 0.f;
    for (int c = 0; c < 64; ++c) { f[c] = htr[c]; g += f[c] * s_wg[c]; }
    g = 1.f / (1.f + __expf(-(g + b_gate[0])));
    for (int c = 0; c < 64; ++c) f[c] = g * f[c] + (1.f - g) * hsr[c];
    for (int hor = 0; hor < 12; ++hor) {
        float s = b_out[hor];
        for (int c = 0; c < 64; ++c) s += f[c] * s_wo[c * 12 + hor];
        out[(size_t)hor * 8000 + i] = s;
    }
}

// ---------------------------------------------------------------------------
extern "C" void kernel_launch(void* const* d_in, const int* in_sizes, int n_in,
                              void* d_out, int out_size, void* d_ws, size_t ws_size,
                              hipStream_t stream)
{
    const float* inputs = (const float*)d_in[0];
    const float* feas   = (const float*)d_in[1];
    const int*   nidx   = (const int*)d_in[2];
    const float* w3  = (const float*)d_in[3];  const float* b3  = (const float*)d_in[4];
    const float* w4  = (const float*)d_in[5];  const float* b4  = (const float*)d_in[6];
    const float* w5  = (const float*)d_in[7];  const float* b5  = (const float*)d_in[8];
    const float* wnh = (const float*)d_in[9];  const float* bnh = (const float*)d_in[10];
    const float* wq  = (const float*)d_in[11]; const float* wk  = (const float*)d_in[12];
    const float* win = (const float*)d_in[13]; const float* bin = (const float*)d_in[14];
    const float* sws = (const float*)d_in[15]; const float* sbs = (const float*)d_in[16];
    const float* swn = (const float*)d_in[17]; const float* sbn = (const float*)d_in[18];
    const float* tw1 = (const float*)d_in[19]; const float* tb1 = (const float*)d_in[20];
    const float* tw2 = (const float*)d_in[21]; const float* tb2 = (const float*)d_in[22];
    const float* wg  = (const float*)d_in[23]; const float* bg  = (const float*)d_in[24];
    const float* wo  = (const float*)d_in[25]; const float* bo  = (const float*)d_in[26];

    // d_out: outputs (96000) | mid_output (6,144,000) | w_adj (64,000)
    float* out  = (float*)d_out;
    float* mid  = out + 96000;
    float* wadj = mid + 6144000;

    // workspace (floats): q, kk, nh, then 3 ping-pong activation buffers
    float* ws = (float*)d_ws;
    float* q  = ws;
    float* kk = q  + 128000;
    float* nh = kk + 128000;
    float* h0 = nh + 128000;
    float* h1 = h0 + 6144000;
    float* b2 = h1 + 6144000;

    embed_kernel<<<8, 256, 0, stream>>>(feas, w3, b3, w4, b4, w5, b5, wq, wk, wnh, bnh, q, kk, nh);
    attn_kernel<<<250, 256, 0, stream>>>(q, kk, nidx, wadj);
    fuse_kernel<<<24000, 256, 0, stream>>>(inputs, win, bin, nh, h0);

    sage_kernel<<<1500, 128, 0, stream>>>(h0, h1, sws,        sbs,      swn,        sbn,      wadj, nidx);
    sage_kernel<<<1500, 128, 0, stream>>>(h1, mid, sws + 4096, sbs + 64, swn + 4096, sbn + 64, wadj, nidx);

    transpose_kernel<<<24000, 256, 0, stream>>>(mid, h0);

    // TCN block 0 (dil=1): h0 -> h1 -> b2 (residual h0)
    tcn_conv_kernel<<<1500, 128, 0, stream>>>(h0, h1, tw1,        tb1,      1, nullptr);
    tcn_conv_kernel<<<1500, 128, 0, stream>>>(h1, b2, tw2,        tb2,      1, h0);
    // TCN block 1 (dil=2): b2 -> h1 -> h0 (residual b2)
    tcn_conv_kernel<<<1500, 128, 0, stream>>>(b2, h1, tw1 + 8192, tb1 + 64, 2, nullptr);
    tcn_conv_kernel<<<1500, 128, 0, stream>>>(h1, h0, tw2 + 8192, tb2 + 64, 2, b2);

    head_kernel<<<32, 256, 0, stream>>>(h0, mid, wg, bg, wo, bo, out);

    (void)in_sizes; (void)n_in; (void)out_size; (void)ws_size;
}